// LinkGNN_20753281974465
// MI455X (gfx1250) — compile-verified
//
#include <hip/hip_runtime.h>
#include <hip/hip_bf16.h>

typedef __attribute__((ext_vector_type(2))) float v2f;
typedef __attribute__((ext_vector_type(8))) float v8f;

// ---------------------------------------------------------------------------
// Utility: zero a float buffer
// ---------------------------------------------------------------------------
__global__ void zero_f32_kernel(float* __restrict__ p, long long n) {
    long long t = (long long)blockIdx.x * blockDim.x + threadIdx.x;
    if (t < n) p[t] = 0.0f;
}

// ---------------------------------------------------------------------------
// Degree count: deg[dst[e]] += 1
// ---------------------------------------------------------------------------
__global__ void degree_kernel(const long long* __restrict__ dst,
                              float* __restrict__ deg, long long E) {
    long long e = (long long)blockIdx.x * blockDim.x + threadIdx.x;
    if (e < E) {
        long long d = dst[e];
        atomicAdd(&deg[d], 1.0f);
    }
}

// dinv[n] = rsqrt(deg[n] + 1)   (in place on deg buffer)
__global__ void dinv_kernel(float* __restrict__ deg, long long N) {
    long long n = (long long)blockIdx.x * blockDim.x + threadIdx.x;
    if (n < N) deg[n] = __frsqrt_rn(deg[n] + 1.0f);
}

// ---------------------------------------------------------------------------
// FP32 WMMA GEMM: C[M x Nf] = A[M x K] * B[K x Nf]
// One wave32 computes one 16x16 tile via V_WMMA_F32_16X16X4_F32.
// blockDim = (32, Nf/16): each y-slice is one wave handling one N tile.
//
// ISA lane layouts (CDNA5 ISA §7.12.2):
//   A 16x4 (MxK):  lane l -> m = l%16;  VGPR0/1 hold K = 2*(l/16), 2*(l/16)+1
//   B 4x16 (KxN):  lane l -> n = l%16;  VGPR0/1 hold K = 2*(l/16), 2*(l/16)+1
//   C/D 16x16:     lane l -> n = l%16;  VGPR i holds row m = 8*(l/16) + i
// ---------------------------------------------------------------------------
__global__ __launch_bounds__(128)
void gemm_wmma_f32_kernel(const float* __restrict__ A,
                          const float* __restrict__ B,
                          float* __restrict__ C,
                          int M, int K, int Nf) {
    const int lane  = threadIdx.x;          // 0..31
    const int ntile = threadIdx.y;          // 0..Nf/16-1
    const int mtile = blockIdx.x;

    const int m  = lane & 15;
    const int n  = lane & 15;
    const int kb = (lane >> 4) << 1;        // 0 or 2
    const int hi = (lane >> 4) << 3;        // 0 or 8 (C/D row base)

    int arow = mtile * 16 + m;
    if (arow >= M) arow = M - 1;            // clamp (no divergence; WMMA needs EXEC all-1)
    const float* __restrict__ ap = A + (size_t)arow * K;
    const float* __restrict__ bp = B + (size_t)(ntile * 16 + n);

    v8f c = {0.f, 0.f, 0.f, 0.f, 0.f, 0.f, 0.f, 0.f};

    for (int k0 = 0; k0 < K; k0 += 4) {
        v2f a, b;
        a.x = ap[k0 + kb];
        a.y = ap[k0 + kb + 1];
        b.x = bp[(size_t)(k0 + kb) * Nf];
        b.y = bp[(size_t)(k0 + kb + 1) * Nf];
        // D = A*B + C ; 8 args: (neg_a, A, neg_b, B, c_mod, C, reuse_a, reuse_b)
        c = __builtin_amdgcn_wmma_f32_16x16x4_f32(
                false, a, false, b, (short)0, c, false, false);
    }

    float* __restrict__ cp = C + (size_t)(mtile * 16) * Nf + ntile * 16 + n;
#pragma unroll
    for (int i = 0; i < 8; ++i) {
        int orow = mtile * 16 + hi + i;
        if (orow < M) cp[(size_t)(hi + i) * Nf] = c[i];
    }
}

// ---------------------------------------------------------------------------
// Edge scatter: agg[dst] += h[src] * dinv[src]*dinv[dst]
// One thread per (edge, 4-feature chunk); F = 64 features.
// ---------------------------------------------------------------------------
__global__ void scatter_kernel(const float* __restrict__ h,
                               const float* __restrict__ dinv,
                               const long long* __restrict__ src,
                               const long long* __restrict__ dst,
                               float* __restrict__ agg, long long E) {
    long long t = (long long)blockIdx.x * blockDim.x + threadIdx.x;
    long long total = E << 4;               // E * 16 chunks
    if (t >= total) return;
    long long e  = t >> 4;
    int       f4 = (int)(t & 15) << 2;      // feature start (0,4,...,60)

    long long s = src[e];
    long long d = dst[e];
    float nrm = dinv[s] * dinv[d];

    const float4 v = *(const float4*)(h + (s << 6) + f4);
    float* __restrict__ ap = agg + (d << 6) + f4;
    atomicAdd(ap + 0, v.x * nrm);
    atomicAdd(ap + 1, v.y * nrm);
    atomicAdd(ap + 2, v.z * nrm);
    atomicAdd(ap + 3, v.w * nrm);
}

// ---------------------------------------------------------------------------
// Combine: agg[n][f] = (agg[n][f] + hpre[n][f]*dinv[n]^2 + bias[f]) [relu?]
// ---------------------------------------------------------------------------
__global__ void combine_kernel(float* __restrict__ agg,
                               const float* __restrict__ hpre,
                               const float* __restrict__ dinv,
                               const float* __restrict__ bias,
                               long long N, int do_relu) {
    long long t = (long long)blockIdx.x * blockDim.x + threadIdx.x;
    if (t >= (N << 6)) return;
    long long n = t >> 6;
    int       f = (int)(t & 63);
    float di = dinv[n];
    float v  = agg[t] + hpre[t] * (di * di) + bias[f];
    if (do_relu) v = fmaxf(v, 0.0f);
    agg[t] = v;
}

// ---------------------------------------------------------------------------
// Decode: out[l] = dot(z[a_l], z[b_l]) over 64 features
// ---------------------------------------------------------------------------
__global__ void decode_kernel(const float* __restrict__ z,
                              const long long* __restrict__ ia,
                              const long long* __restrict__ ib,
                              float* __restrict__ out, long long L) {
    long long t = (long long)blockIdx.x * blockDim.x + threadIdx.x;
    if (t >= L) return;
    const float4* za = (const float4*)(z + (ia[t] << 6));
    const float4* zb = (const float4*)(z + (ib[t] << 6));
    float s = 0.0f;
#pragma unroll
    for (int i = 0; i < 16; ++i) {
        float4 va = za[i];
        float4 vb = zb[i];
        s += va.x * vb.x + va.y * vb.y + va.z * vb.z + va.w * vb.w;
    }
    out[t] = s;
}

// ---------------------------------------------------------------------------
// Host-side launcher
// ---------------------------------------------------------------------------
extern "C" void kernel_launch(void* const* d_in, const int* in_sizes, int n_in,
                              void* d_out, int out_size, void* d_ws, size_t ws_size,
                              hipStream_t stream) {
    const float*     x    = (const float*)d_in[0];
    const float*     W1   = (const float*)d_in[1];
    const float*     b1   = (const float*)d_in[2];
    const float*     W2   = (const float*)d_in[3];
    const float*     b2   = (const float*)d_in[4];
    const long long* ei   = (const long long*)d_in[5];   // (2, E) int64
    const long long* eli  = (const long long*)d_in[6];   // (2, L) int64
    float*           out  = (float*)d_out;

    const int H   = in_sizes[2];                // 64
    const int INF = in_sizes[1] / H;            // 128
    const long long N = (long long)in_sizes[0] / INF;   // 100000
    const long long E = (long long)in_sizes[5] / 2;     // 1200000
    const long long L = (long long)in_sizes[6] / 2;     // 200000

    const long long* src = ei;          // row 0
    const long long* dst = ei + E;      // row 1
    const long long* la  = eli;
    const long long* lb  = eli + L;

    // Workspace layout (floats, all 16B aligned):
    //   dinv : N
    //   h0   : N*64   (layer-1 pre-agg; later reused for layer-2 pre-agg h1)
    //   agg  : N*64   (layer-1 agg -> relu'd h, in place)
    //   z    : N*64   (layer-2 agg -> z, in place)
    float* dinv = (float*)d_ws;
    float* h0   = dinv + N;
    float* agg  = h0 + N * 64;
    float* z    = agg + N * 64;

    const int TB = 256;
    auto blocks = [&](long long n) { return (unsigned)((n + TB - 1) / TB); };

    // 1) degrees -> dinv
    zero_f32_kernel<<<blocks(N), TB, 0, stream>>>(dinv, N);
    degree_kernel<<<blocks(E), TB, 0, stream>>>(dst, dinv, E);
    dinv_kernel<<<blocks(N), TB, 0, stream>>>(dinv, N);

    // 2) h0 = x @ W1   (M=N, K=128, Nf=64) — FP32 WMMA
    {
        dim3 blk(32, H / 16);
        dim3 grd((unsigned)((N + 15) / 16));
        gemm_wmma_f32_kernel<<<grd, blk, 0, stream>>>(x, W1, h0, (int)N, INF, H);
    }

    // 3) layer-1 aggregation + self loop + bias + relu  (h lives in `agg`)
    zero_f32_kernel<<<blocks(N * 64), TB, 0, stream>>>(agg, N * 64);
    scatter_kernel<<<blocks(E * 16), TB, 0, stream>>>(h0, dinv, src, dst, agg, E);
    combine_kernel<<<blocks(N * 64), TB, 0, stream>>>(agg, h0, dinv, b1, N, 1);

    // 4) h1 = h @ W2   (M=N, K=64, Nf=64) into h0 buffer — FP32 WMMA
    {
        dim3 blk(32, H / 16);
        dim3 grd((unsigned)((N + 15) / 16));
        gemm_wmma_f32_kernel<<<grd, blk, 0, stream>>>(agg, W2, h0, (int)N, H, H);
    }

    // 5) layer-2 aggregation + self loop + bias (no relu)  (z lives in `z`)
    zero_f32_kernel<<<blocks(N * 64), TB, 0, stream>>>(z, N * 64);
    scatter_kernel<<<blocks(E * 16), TB, 0, stream>>>(h0, dinv, src, dst, z, E);
    combine_kernel<<<blocks(N * 64), TB, 0, stream>>>(z, h0, dinv, b2, N, 0);

    // 6) decode label edges
    decode_kernel<<<blocks(L), TB, 0, stream>>>(z, la, lb, out, L);
}